// AgentLoss_13898514169999
// MI455X (gfx1250) — compile-verified
//
#include <hip/hip_runtime.h>
#include <hip/hip_bf16.h>

typedef float v2f __attribute__((ext_vector_type(2)));
typedef float v8f __attribute__((ext_vector_type(8)));

#define T_LEN   16384
#define C_LEN   2048
#define NTH     256
#define CHUNK   (T_LEN / NTH)   // 64

// ---------------------------------------------------------------------------
// Kernel A: parallel affine-recurrence scan for the A3C loss.
// Rc_t = g*Rc_{t+1} + r_t   (Rc_T = R0)
// gae_t = g*gae_{t+1} + delta_t, delta_t = r_t + g*v_{t+1} - v_t
// Each thread owns a contiguous 64-step chunk; chunk transform is affine
// (a = g^64, b = local discounted sum), composed sequentially by thread 0.
// ---------------------------------------------------------------------------
__global__ void a3c_scan_kernel(const float* __restrict__ Rp,
                                const float* __restrict__ gp,
                                const float* __restrict__ rew,
                                const float* __restrict__ lp,
                                const float* __restrict__ val,   // T+1 elems
                                const float* __restrict__ ent,
                                float* __restrict__ ws) {
    __shared__ float sA[NTH], sBR[NTH], sBG[NTH];
    __shared__ float cR[NTH], cG[NTH];
    __shared__ float redP[NTH], redV[NTH];

    const int tid = threadIdx.x;
    const float g = gp[0];
    const int lo = tid * CHUNK;
    const int hi = lo + CHUNK;

    // Phase 1: local (exact sequential) chunk summaries, carry-in = 0
    float bR = 0.f, bG = 0.f, a = 1.f;
    for (int t = hi - 1; t >= lo; --t) {
        const float rt = rew[t];
        bR = rt + g * bR;
        const float delta = rt + g * val[t + 1] - val[t];
        bG = delta + g * bG;
        a *= g;
    }
    sA[tid] = a; sBR[tid] = bR; sBG[tid] = bG;
    __syncthreads();

    // Phase 2: compose chunk transforms right-to-left (carry entering chunk i)
    if (tid == 0) {
        cR[NTH - 1] = Rp[0];
        cG[NTH - 1] = 0.f;
        for (int i = NTH - 2; i >= 0; --i) {
            cR[i] = sBR[i + 1] + sA[i + 1] * cR[i + 1];
            cG[i] = sBG[i + 1] + sA[i + 1] * cG[i + 1];
        }
    }
    __syncthreads();

    // Phase 3: replay chunk with exact carries, accumulate losses
    float Rc = cR[tid], gae = cG[tid];
    float pl = 0.f, vl = 0.f;
    for (int t = hi - 1; t >= lo; --t) {
        const float rt = rew[t];
        Rc = g * Rc + rt;
        const float adv = Rc - val[t];
        vl += 0.5f * adv * adv;
        const float delta = rt + g * val[t + 1] - val[t];
        gae = g * gae + delta;
        pl += -lp[t] * gae - 0.01f * ent[t];
    }
    redP[tid] = pl; redV[tid] = vl;
    __syncthreads();
    for (int off = NTH / 2; off > 0; off >>= 1) {
        if (tid < off) { redP[tid] += redP[tid + off]; redV[tid] += redV[tid + off]; }
        __syncthreads();
    }
    if (tid == 0) ws[0] = redP[0] + 0.5f * redV[0];   // a3c loss
}

// ---------------------------------------------------------------------------
// Kernel B: per-row logsumexp - picked. 128 MB of reads -> HBM-bandwidth bound
// (~5.5 us at 23.3 TB/s). One 256-thread block per row, float4 coalesced loads.
// ---------------------------------------------------------------------------
__global__ void opponent_row_kernel(const float* __restrict__ opp,
                                    const int* __restrict__ gt,
                                    float* __restrict__ rowOut) {
    __shared__ float s[NTH];
    const int row = blockIdx.x;
    const int tid = threadIdx.x;
    const float* rp = opp + (size_t)row * C_LEN;
    const float4* rp4 = (const float4*)rp;

    const float4 x0 = rp4[tid * 2];
    const float4 x1 = rp4[tid * 2 + 1];

    float m = fmaxf(fmaxf(fmaxf(x0.x, x0.y), fmaxf(x0.z, x0.w)),
                    fmaxf(fmaxf(x1.x, x1.y), fmaxf(x1.z, x1.w)));
    s[tid] = m; __syncthreads();
    for (int off = NTH / 2; off > 0; off >>= 1) {
        if (tid < off) s[tid] = fmaxf(s[tid], s[tid + off]);
        __syncthreads();
    }
    const float rm = s[0];
    __syncthreads();

    float sum = expf(x0.x - rm) + expf(x0.y - rm) + expf(x0.z - rm) + expf(x0.w - rm)
              + expf(x1.x - rm) + expf(x1.y - rm) + expf(x1.z - rm) + expf(x1.w - rm);
    s[tid] = sum; __syncthreads();
    for (int off = NTH / 2; off > 0; off >>= 1) {
        if (tid < off) s[tid] += s[tid + off];
        __syncthreads();
    }
    if (tid == 0)
        rowOut[row] = rm + logf(s[0]) - rp[gt[row]];
}

// ---------------------------------------------------------------------------
// Kernel C: deterministic 16384-element reduction via V_WMMA_F32_16X16X4_F32.
// B = ones(4x16)  =>  D[m,n] = sum_k A[m,k] + C[m,n]; each WMMA folds 64
// values into the f32 accumulator. One full wave32, EXEC all ones.
// D layout: lanes 0-15 VGPR j = rowsum(M=j); lanes 16-31 VGPR j = rowsum(M=8+j)
// => total = half0 + half1 via shfl_xor(16).
// ---------------------------------------------------------------------------
__global__ void reduce_wmma_kernel(const float* __restrict__ rows,
                                   const float* __restrict__ a3c,
                                   float* __restrict__ out) {
    const int lane = threadIdx.x;            // 0..31, full wave
    const float2* p = (const float2*)rows;   // 2 floats per lane per iter

    v8f acc = {};
    v2f ones; ones[0] = 1.0f; ones[1] = 1.0f;

    for (int it = 0; it < T_LEN / 64; ++it) {         // 256 WMMAs
        const float2 d = p[it * 32 + lane];
        v2f av; av[0] = d.x; av[1] = d.y;
        acc = __builtin_amdgcn_wmma_f32_16x16x4_f32(
            /*neg_a=*/false, av, /*neg_b=*/false, ones,
            /*c_mod=*/(short)0, acc, /*reuse_a=*/false, /*reuse_b=*/false);
    }

    float partial = acc[0] + acc[1] + acc[2] + acc[3]
                  + acc[4] + acc[5] + acc[6] + acc[7];
    const float total = partial + __shfl_xor(partial, 16, 32);

    if (lane == 0)
        out[0] = a3c[0] + total * (1.0f / (float)T_LEN);
}

// ---------------------------------------------------------------------------
extern "C" void kernel_launch(void* const* d_in, const int* in_sizes, int n_in,
                              void* d_out, int out_size, void* d_ws, size_t ws_size,
                              hipStream_t stream) {
    (void)in_sizes; (void)n_in; (void)out_size; (void)ws_size;

    const float* R     = (const float*)d_in[0];
    const float* gamma = (const float*)d_in[1];
    const float* rew   = (const float*)d_in[2];
    const float* lp    = (const float*)d_in[3];
    const float* val   = (const float*)d_in[4];
    const float* ent   = (const float*)d_in[5];
    const float* opp   = (const float*)d_in[6];
    const int*   gt    = (const int*)d_in[7];

    float* ws     = (float*)d_ws;
    float* rowOut = ws + 64;                  // 16384 per-row losses

    a3c_scan_kernel<<<1, NTH, 0, stream>>>(R, gamma, rew, lp, val, ent, ws);
    opponent_row_kernel<<<T_LEN, NTH, 0, stream>>>(opp, gt, rowOut);
    reduce_wmma_kernel<<<1, 32, 0, stream>>>(rowOut, ws, (float*)d_out);
}